// SchemaHeteroGAT_65747359367347
// MI455X (gfx1250) — compile-verified
//
#include <hip/hip_runtime.h>
#include <cstddef>

#define NHEAD 4
#define DHEAD 64
#define FH    256
#define NT_N  20000
#define NC_N  200000
#define NF_N  40000
#define E_HC_N 200000
#define E_BT_N 200000
#define E_IS_N 80000
#define E_PT_N 80000

typedef __attribute__((ext_vector_type(2))) float v2f;
typedef __attribute__((ext_vector_type(8))) float v8f;

// ---------------------------------------------------------------------------
// float atomic max via integer ordering trick (init must be -inf)
// ---------------------------------------------------------------------------
__device__ __forceinline__ void atomic_max_f32(float* addr, float val) {
  if (val >= 0.0f)
    atomicMax((int*)addr, __float_as_int(val));
  else
    atomicMin((unsigned int*)addr, __float_as_uint(val));
}

__device__ __forceinline__ float elu1(float x) {
  return (x > 0.0f) ? x : (expf(x) - 1.0f);
}

// ---------------------------------------------------------------------------
// WMMA f32 GEMM: C[M,N] = act(A[M,K] @ W[K,N] + bias[N])
// one wave -> 16 rows x 64 cols (4 tiles, A fragment reused across B tiles).
// block = 256 threads = 8 waves = 128 rows; grid = (ceil(M/128), N/64).
// W block-column (K x 64) is staged in LDS pair-packed so each B fragment is
// one ds_load_b64:  element (k, n) -> Wl[(k>>1)*128 + 2n + (k&1)].
// A rows are clamped (not predicated): garbage rows never stored.
// K in {64, 256}; N multiple of 64.  ACT: 0 = none, 1 = leaky_relu(0.01)
// ---------------------------------------------------------------------------
template <int ACT>
__global__ __launch_bounds__(256) void gemm_wmma_f32(
    const float* __restrict__ A, const float* __restrict__ W,
    const float* __restrict__ bias, float* __restrict__ C,
    int M, int K, int N) {
  __shared__ float Wl[16384];  // up to 256x64 floats, pair-packed (64 KB)

  const int tid  = threadIdx.x;
  const int lane = tid & 31;
  const int wave = tid >> 5;
  const int m0 = (blockIdx.x * 8 + wave) * 16;
  const int n0 = blockIdx.y * 64;
  const int half = lane >> 4;     // 0: K pair {k,k+1}; 1: {k+2,k+3}
  const int l16  = lane & 15;

  int arow = m0 + l16;
  if (arow >= M) arow = M - 1;    // clamp: extra rows computed, never stored
  const float* Arowp = A + (size_t)arow * K;
  __builtin_prefetch(Arowp, 0, 1);  // warm A while W staging proceeds

  // stage W(:, n0:n0+64) into LDS, pair-packed
  const int KN = K << 6;
  for (int idx = tid; idx < KN; idx += 256) {
    const int kk = idx >> 6, nn = idx & 63;
    Wl[((kk >> 1) << 7) + (nn << 1) + (kk & 1)] = W[(size_t)kk * N + n0 + nn];
  }
  __syncthreads();

  if (m0 >= M) return;  // wave-uniform; no barriers after this point

  v8f acc0 = {}, acc1 = {}, acc2 = {}, acc3 = {};
  const v2f* Arow = (const v2f*)Arowp;   // 8B aligned (row stride 64/256 floats)
  const v2f* Wp   = (const v2f*)Wl;

  for (int k0 = 0; k0 < K; k0 += 4) {
    const int p = (k0 >> 1) + half;      // K-pair index for this lane half
    const v2f a  = Arow[p];
    const v2f b0 = Wp[p * 64 + l16];
    const v2f b1 = Wp[p * 64 + 16 + l16];
    const v2f b2 = Wp[p * 64 + 32 + l16];
    const v2f b3 = Wp[p * 64 + 48 + l16];
    acc0 = __builtin_amdgcn_wmma_f32_16x16x4_f32(false, a, false, b0, (short)0, acc0, false, false);
    acc1 = __builtin_amdgcn_wmma_f32_16x16x4_f32(false, a, false, b1, (short)0, acc1, false, false);
    acc2 = __builtin_amdgcn_wmma_f32_16x16x4_f32(false, a, false, b2, (short)0, acc2, false, false);
    acc3 = __builtin_amdgcn_wmma_f32_16x16x4_f32(false, a, false, b3, (short)0, acc3, false, false);
  }

  const int mbase = m0 + (half << 3);    // lanes 16-31 hold rows +8
  #pragma unroll
  for (int t = 0; t < 4; ++t) {
    v8f acc;
    if      (t == 0) acc = acc0;
    else if (t == 1) acc = acc1;
    else if (t == 2) acc = acc2;
    else             acc = acc3;
    const int n = n0 + t * 16 + l16;
    const float bv = bias ? bias[n] : 0.0f;
    #pragma unroll
    for (int r = 0; r < 8; ++r) {
      const int m = mbase + r;
      if (m < M) {
        float v = acc[r] + bv;
        if (ACT == 1) v = (v > 0.0f) ? v : 0.01f * v;
        C[(size_t)m * N + n] = v;
      }
    }
  }
}

// ---------------------------------------------------------------------------
// init per-(dst,head) running max (-inf) and denominator (0)
// ---------------------------------------------------------------------------
__global__ void init_md(float* __restrict__ m, float* __restrict__ den, int n) {
  int i = blockIdx.x * blockDim.x + threadIdx.x;
  if (i < n) { m[i] = -__builtin_inff(); den[i] = 0.0f; }
}

// ---------------------------------------------------------------------------
// per-edge logits: logit[e,h] = sum_d leaky0.2(xl[src]+xr[dst]) * att[h,d]
// one wave per edge; lane covers channels [lane*8, lane*8+8) -> head = lane>>3
// (uniform per lane); segmented 8-lane shuffle reduction; b128 loads.
// ---------------------------------------------------------------------------
__global__ __launch_bounds__(256) void edge_logit_kernel(
    const float* __restrict__ xl, const float* __restrict__ xr,
    const int* __restrict__ src, const int* __restrict__ dst,
    const float* __restrict__ att, float* __restrict__ logit,
    float* __restrict__ mbuf, int E) {
  const int e = blockIdx.x * 8 + (threadIdx.x >> 5);
  const int lane = threadIdx.x & 31;
  if (e >= E) return;
  const int s = src[e], d = dst[e];
  const float4* pl = (const float4*)(xl + (size_t)s * FH) + lane * 2;
  const float4* pr = (const float4*)(xr + (size_t)d * FH) + lane * 2;
  const float4* pa = (const float4*)att + lane * 2;
  float acc = 0.0f;
  #pragma unroll
  for (int j = 0; j < 2; ++j) {
    const float4 lv = pl[j], rv = pr[j], av = pa[j];
    float v;
    v = lv.x + rv.x; v = (v > 0.0f) ? v : 0.2f * v; acc += v * av.x;
    v = lv.y + rv.y; v = (v > 0.0f) ? v : 0.2f * v; acc += v * av.y;
    v = lv.z + rv.z; v = (v > 0.0f) ? v : 0.2f * v; acc += v * av.z;
    v = lv.w + rv.w; v = (v > 0.0f) ? v : 0.2f * v; acc += v * av.w;
  }
  acc += __shfl_xor(acc, 4, 32);
  acc += __shfl_xor(acc, 2, 32);
  acc += __shfl_xor(acc, 1, 32);
  if ((lane & 7) == 0) {
    const int h = lane >> 3;
    logit[e * 4 + h] = acc;
    atomic_max_f32(&mbuf[d * 4 + h], acc);
  }
}

// ---------------------------------------------------------------------------
// z = exp(logit - m[dst]); den[dst] += z   (in-place over logit buffer)
// ---------------------------------------------------------------------------
__global__ void edge_z_kernel(const int* __restrict__ dst, const float* __restrict__ mbuf,
                              float* __restrict__ zbuf, float* __restrict__ den, int E) {
  const int i = blockIdx.x * blockDim.x + threadIdx.x;
  if (i >= E * 4) return;
  const int e = i >> 2, h = i & 3;
  const int d = dst[e];
  const float z = expf(zbuf[i] - mbuf[d * 4 + h]);
  zbuf[i] = z;
  atomicAdd(&den[d * 4 + h], z);
}

// ---------------------------------------------------------------------------
// out[dst, c] += xl[src, c] * alpha[c>>6]; 64 threads/edge, float4 per thread
// ---------------------------------------------------------------------------
__global__ __launch_bounds__(256) void edge_scatter_kernel(
    const float* __restrict__ xl, const int* __restrict__ src, const int* __restrict__ dst,
    const float* __restrict__ z, const float* __restrict__ den,
    float* __restrict__ out, int E) {
  const int e = blockIdx.x * 4 + (threadIdx.x >> 6);
  if (e >= E) return;
  const int t = threadIdx.x & 63;      // covers channels [4t, 4t+4)
  const int s = src[e], d = dst[e];
  const int h = t >> 4;
  const float alpha = z[e * 4 + h] / den[d * 4 + h];
  const float4 xv = *((const float4*)(xl + (size_t)s * FH) + t);
  float* op = out + (size_t)d * FH + t * 4;
  atomicAdd(op + 0, xv.x * alpha);
  atomicAdd(op + 1, xv.y * alpha);
  atomicAdd(op + 2, xv.z * alpha);
  atomicAdd(op + 3, xv.w * alpha);
}

// ---------------------------------------------------------------------------
// next = elu(scale * (acc + b1 [+ b2])), float4-vectorized (row = 64 float4)
// ---------------------------------------------------------------------------
__global__ void finalize_kernel(const float4* __restrict__ acc, const float4* __restrict__ b1,
                                const float4* __restrict__ b2, float scale,
                                float4* __restrict__ out, int total4) {
  const int i = blockIdx.x * blockDim.x + threadIdx.x;
  if (i >= total4) return;
  const int c = i & 63;
  float4 v = acc[i];
  const float4 bb = b1[c];
  v.x += bb.x; v.y += bb.y; v.z += bb.z; v.w += bb.w;
  if (b2) {
    const float4 b2v = b2[c];
    v.x += b2v.x; v.y += b2v.y; v.z += b2v.z; v.w += b2v.w;
  }
  v.x = elu1(v.x * scale); v.y = elu1(v.y * scale);
  v.z = elu1(v.z * scale); v.w = elu1(v.w * scale);
  out[i] = v;
}

// ---------------------------------------------------------------------------
extern "C" void kernel_launch(void* const* d_in, const int* in_sizes, int n_in,
                              void* d_out, int out_size, void* d_ws, size_t ws_size,
                              hipStream_t stream) {
  const float* x_table  = (const float*)d_in[0];
  const float* x_column = (const float*)d_in[1];
  const float* x_fk     = (const float*)d_in[2];
  const float* lin_w    = (const float*)d_in[3];
  const float* lin_b    = (const float*)d_in[4];
  const float* out_w    = (const float*)d_in[5];
  const float* out_b    = (const float*)d_in[6];
  const float* Wl[2]   = {(const float*)d_in[7],  (const float*)d_in[11]};
  const float* Wr[2]   = {(const float*)d_in[8],  (const float*)d_in[12]};
  const float* attp[2] = {(const float*)d_in[9],  (const float*)d_in[13]};
  const float* bp[2]   = {(const float*)d_in[10], (const float*)d_in[14]};
  const int* src_e[4] = {(const int*)d_in[15], (const int*)d_in[17],
                         (const int*)d_in[19], (const int*)d_in[21]};
  const int* dst_e[4] = {(const int*)d_in[16], (const int*)d_in[18],
                         (const int*)d_in[20], (const int*)d_in[22]};
  const int Ecnt[4] = {E_HC_N, E_BT_N, E_IS_N, E_PT_N};

  const size_t NALL = (size_t)NT_N + NC_N + NF_N;  // 260000
  float* bufX = (float*)d_ws;                      // features (ping)
  float* bufY = bufX + NALL * FH;                  // accumulators (pong)
  float* xlb  = bufY + NALL * FH;                  // src projections [<=NC, 256]
  float* xrb  = xlb + (size_t)NC_N * FH;           // dst projections [<=NC, 256]
  float* zb   = xrb + (size_t)NC_N * FH;           // logits / z  [<=E, 4]
  float* mb   = zb + (size_t)NC_N * 4;             // segment max [<=Nd, 4]
  float* db   = mb + (size_t)NC_N * 4;             // segment sum [<=Nd, 4]

  auto gemm = [&](const float* A, const float* W, const float* bias, float* C,
                  int M, int K, int N, int act) {
    dim3 grid((M + 127) / 128, N / 64);
    if (act)
      gemm_wmma_f32<1><<<grid, dim3(256), 0, stream>>>(A, W, bias, C, M, K, N);
    else
      gemm_wmma_f32<0><<<grid, dim3(256), 0, stream>>>(A, W, bias, C, M, K, N);
  };

  // ---- input projections: [N,256] @ [256,64] + b, leaky_relu(0.01) ----
  gemm(x_table,  lin_w + 0 * 256 * 64, lin_b + 0 * 64, bufX,                              NT_N, 256, 64, 1);
  gemm(x_column, lin_w + 1 * 256 * 64, lin_b + 1 * 64, bufX + (size_t)NT_N * 64,          NC_N, 256, 64, 1);
  gemm(x_fk,     lin_w + 2 * 256 * 64, lin_b + 2 * 64, bufX + (NT_N + (size_t)NC_N) * 64, NF_N, 256, 64, 1);

  // ---- two hetero-GATv2 layers ----
  for (int l = 0; l < 2; ++l) {
    const int Kin = l ? FH : 64;
    const float* xt = bufX;
    const float* xc = bufX + (size_t)NT_N * Kin;
    const float* xf = bufX + (NT_N + (size_t)NC_N) * Kin;
    const float* sx[4] = {xt, xc, xc, xf};   // hc, bt, is, pt sources
    const float* dx[4] = {xc, xt, xf, xc};   // destinations
    const int Ns[4] = {NT_N, NC_N, NC_N, NF_N};
    const int Nd[4] = {NC_N, NT_N, NF_N, NC_N};
    float* accT = bufY;
    float* accC = bufY + (size_t)NT_N * FH;
    float* accF = bufY + (NT_N + (size_t)NC_N) * FH;
    float* accp[4] = {accC, accT, accF, accC};  // hc & pt both into column acc

    hipMemsetAsync(bufY, 0, NALL * FH * sizeof(float), stream);

    for (int et = 0; et < 4; ++et) {
      const int E = Ecnt[et];
      gemm(sx[et], Wl[l] + (size_t)et * Kin * FH, nullptr, xlb, Ns[et], Kin, FH, 0);
      gemm(dx[et], Wr[l] + (size_t)et * Kin * FH, nullptr, xrb, Nd[et], Kin, FH, 0);
      const int nmd = Nd[et] * 4;
      init_md<<<(nmd + 255) / 256, 256, 0, stream>>>(mb, db, nmd);
      edge_logit_kernel<<<(E + 7) / 8, 256, 0, stream>>>(
          xlb, xrb, src_e[et], dst_e[et], attp[l] + et * 256, zb, mb, E);
      edge_z_kernel<<<(E * 4 + 255) / 256, 256, 0, stream>>>(dst_e[et], mb, zb, db, E);
      edge_scatter_kernel<<<(E + 3) / 4, 256, 0, stream>>>(xlb, src_e[et], dst_e[et],
                                                           zb, db, accp[et], E);
    }

    // finalize: add biases, HeteroConv mean for columns, ELU -> next features
    finalize_kernel<<<(NT_N * 64 + 255) / 256, 256, 0, stream>>>(
        (const float4*)accT, (const float4*)(bp[l] + 1 * FH), (const float4*)nullptr,
        1.0f, (float4*)bufX, NT_N * 64);
    finalize_kernel<<<(NC_N * 64 + 255) / 256, 256, 0, stream>>>(
        (const float4*)accC, (const float4*)(bp[l] + 0 * FH), (const float4*)(bp[l] + 3 * FH),
        0.5f, (float4*)(bufX + (size_t)NT_N * FH), NC_N * 64);
    finalize_kernel<<<(NF_N * 64 + 255) / 256, 256, 0, stream>>>(
        (const float4*)accF, (const float4*)(bp[l] + 2 * FH), (const float4*)nullptr,
        1.0f, (float4*)(bufX + (NT_N + (size_t)NC_N) * FH), NF_N * 64);
  }

  // ---- output projections into d_out (xt | xc | xf, each [N,256]) ----
  float* outp = (float*)d_out;
  gemm(bufX,                              out_w + 0 * 256 * 256, out_b + 0 * 256,
       outp,                              NT_N, 256, 256, 0);
  gemm(bufX + (size_t)NT_N * FH,          out_w + 1 * 256 * 256, out_b + 1 * 256,
       outp + (size_t)NT_N * FH,          NC_N, 256, 256, 0);
  gemm(bufX + (NT_N + (size_t)NC_N) * FH, out_w + 2 * 256 * 256, out_b + 2 * 256,
       outp + (NT_N + (size_t)NC_N) * FH, NF_N, 256, 256, 0);
}